// RebuildCostVolume_13718125543603
// MI455X (gfx1250) — compile-verified
//
#include <hip/hip_runtime.h>
#include <cstdint>

// Residual L1 cost volume with horizontal bilinear warp.
//   out[b,k,h,j] = (0 <= j+4-k < W) ? sum_c |x[b,c,h,j] - y_w[b,c,h,j-(k-4)]| : 0
//   y_w = horizontal bilinear sample of y at (j - disp), zeros padding.
//
// Memory-bound (~310 MB moved, <1 GFLOP). Strategy: one workgroup per (b,h)
// row; stage raw y row into LDS with CDNA5 async global->LDS loads
// (ASYNCcnt / s_wait_asynccnt), warp it once into a halo-padded LDS buffer,
// then each thread produces 2 adjacent output columns for all 9 planes,
// reading x exactly once (coalesced float2) and y_w from LDS (9x reuse).

#define D_PLANES 9
#define B_N 8
#define C_N 32
#define H_N 256
#define W_N 512
#define YW_STRIDE (W_N + 8)   // 4-col halo each side

__global__ __launch_bounds__(256)
void cost_volume_warp_kernel(const float* __restrict__ x,
                             const float* __restrict__ disp,
                             const float* __restrict__ y,
                             float* __restrict__ out)
{
    // 64 KB raw y row [c][w], 65 KB warped row [c][m+4] (halo padded)
    __shared__ __align__(16) float s_y[C_N * W_N];
    __shared__ __align__(16) float s_yw[C_N * YW_STRIDE];

    const int t = (int)threadIdx.x;
    const int h = (int)blockIdx.x;
    const int b = (int)blockIdx.y;

    // Base of y[b,0,h,0]; channels are H*W floats apart.
    const float* yrow = y + (size_t)b * C_N * H_N * W_N + (size_t)h * W_N;

    // ---- Stage 1: async-copy y row (32 strided 2KB chunks) into s_y -------
    // 32 ch * 128 x 16B chunks = 4096 chunk loads; 16 per lane.
    {
        const uint32_t lds_y_base = (uint32_t)(uintptr_t)&s_y[0]; // low 32 bits = LDS byte addr
        const uint32_t col16 = (uint32_t)(t & 127);               // 16B chunk within channel
        const uint32_t cstart = (uint32_t)(t >> 7);               // 0 or 1
#pragma unroll
        for (int i = 0; i < 16; ++i) {
            uint32_t c     = cstart + 2u * (uint32_t)i;
            uint32_t goff  = c * (uint32_t)(H_N * W_N * 4) + col16 * 16u; // byte offset vs yrow
            uint32_t laddr = lds_y_base + c * (uint32_t)(W_N * 4) + col16 * 16u;
            asm volatile("global_load_async_to_lds_b128 %0, %1, %2"
                         :: "v"(laddr), "v"(goff), "s"(yrow)
                         : "memory");
        }
    }

    // ---- Overlap: zero the y_w halo and compute warp coords ---------------
    {   // 8 halo entries per channel * 32 channels = 256 threads
        int c = t >> 3, e = t & 7;
        int idx = c * YW_STRIDE + ((e < 4) ? e : (W_N + e));   // m=-4..-1 and m=512..515
        s_yw[idx] = 0.0f;
    }

    const int j0 = 2 * t;   // this thread owns columns j0, j0+1
    const float2 dv = *(const float2*)(disp + ((size_t)b * H_N + h) * W_N + j0);
    float sx0 = (float)j0 - dv.x;
    float sx1 = (float)(j0 + 1) - dv.y;
    float f0 = floorf(sx0), f1 = floorf(sx1);
    float wx0 = sx0 - f0,   wx1 = sx1 - f1;
    int  i00 = (int)f0,     i10 = (int)f1;
    bool a0v = (i00     >= 0) && (i00     < W_N);
    bool b0v = (i00 + 1 >= 0) && (i00 + 1 < W_N);
    bool a1v = (i10     >= 0) && (i10     < W_N);
    bool b1v = (i10 + 1 >= 0) && (i10 + 1 < W_N);
    int a0 = min(max(i00,     0), W_N - 1);
    int b0 = min(max(i00 + 1, 0), W_N - 1);
    int a1 = min(max(i10,     0), W_N - 1);
    int b1 = min(max(i10 + 1, 0), W_N - 1);

    // Wait for this wave's async LDS fills, then workgroup barrier.
    asm volatile("s_wait_asynccnt 0x0" ::: "memory");
    __syncthreads();

    // ---- Stage 2: warp y row into s_yw (column m stored at [c][m+4]) ------
#pragma unroll 4
    for (int c = 0; c < C_N; ++c) {
        const float* yc = &s_y[c * W_N];
        float g00 = a0v ? yc[a0] : 0.0f;
        float g01 = b0v ? yc[b0] : 0.0f;
        float g10 = a1v ? yc[a1] : 0.0f;
        float g11 = b1v ? yc[b1] : 0.0f;
        float* ywc = &s_yw[c * YW_STRIDE + 4];
        ywc[j0]     = g00 * (1.0f - wx0) + g01 * wx0;
        ywc[j0 + 1] = g10 * (1.0f - wx1) + g11 * wx1;
    }
    __syncthreads();

    // ---- Stage 3: 9-plane L1 cost for columns j0, j0+1 --------------------
    const float* xrow = x + (size_t)b * C_N * H_N * W_N + (size_t)h * W_N;

    float acc0[D_PLANES], acc1[D_PLANES];
#pragma unroll
    for (int k = 0; k < D_PLANES; ++k) { acc0[k] = 0.0f; acc1[k] = 0.0f; }

#pragma unroll 4
    for (int c = 0; c < C_N; ++c) {
        float2 xv = *(const float2*)(xrow + (size_t)c * H_N * W_N + j0);
        const float* ywc = &s_yw[c * YW_STRIDE];  // entry i holds column m = i-4
        float win[10];                            // m = j0-4 .. j0+5
#pragma unroll
        for (int i = 0; i < 10; ++i) win[i] = ywc[j0 + i];
#pragma unroll
        for (int k = 0; k < D_PLANES; ++k) {
            acc0[k] += fabsf(xv.x - win[8 - k]);  // m = j0   + 4 - k
            acc1[k] += fabsf(xv.y - win[9 - k]);  // m = j0+1 + 4 - k
        }
    }

    // ---- Write out, masking planes whose shifted column is out of range ---
#pragma unroll
    for (int k = 0; k < D_PLANES; ++k) {
        int m0 = j0 + 4 - k;
        int m1 = j0 + 1 + 4 - k;
        float2 o;
        o.x = (m0 >= 0 && m0 < W_N) ? acc0[k] : 0.0f;
        o.y = (m1 >= 0 && m1 < W_N) ? acc1[k] : 0.0f;
        *(float2*)(out + (((size_t)b * D_PLANES + k) * H_N + h) * W_N + j0) = o;
    }
}

extern "C" void kernel_launch(void* const* d_in, const int* in_sizes, int n_in,
                              void* d_out, int out_size, void* d_ws, size_t ws_size,
                              hipStream_t stream) {
    (void)in_sizes; (void)n_in; (void)d_ws; (void)ws_size; (void)out_size;
    // setup_inputs() dict order: x, disp, y
    const float* x    = (const float*)d_in[0];
    const float* disp = (const float*)d_in[1];
    const float* y    = (const float*)d_in[2];
    float* out = (float*)d_out;

    dim3 grid(H_N, B_N);   // one workgroup per (b, h) row
    dim3 block(256);
    cost_volume_warp_kernel<<<grid, block, 0, stream>>>(x, disp, y, out);
}